// GeneralConv_68736656605428
// MI455X (gfx1250) — compile-verified
//
#include <hip/hip_runtime.h>

// CDNA5 / gfx1250, wave32.
// Grouped graph convolution:
//   Phase 1 (WMMA): x_all[n][27][64]: for each (k,g), [n x 16] * [16 x 16]
//   Phase 2: out[i] += x_all[j, k] per edge; out initialized with bias.
//
// fp32 end-to-end: kernel is HBM/atomic bound (691 MB x_all write ~30us at
// 23.3 TB/s vs 5.5 GFLOP of math) -> V_WMMA_F32_16X16X4_F32 (exact fp32,
// 4 chained K=4 ops per 16-wide reduction) is the right matrix op.

typedef __attribute__((ext_vector_type(2))) float v2f;
typedef __attribute__((ext_vector_type(8))) float v8f;

#define IN_C 64
#define OUT_C 64
#define NK 27
#define NG 4
#define CSTRIDE (16 * NK) /* flat-index step for c+1 in weight[] = 432 */

// ---------------------------------------------------------------------------
// Phase 1 (full tiles only -- completely branch/bounds free):
// one wave = 16-node tile x one group g, loops over k.
// A (16x4 f32): VGPR0 = {c=4q+2*half}, VGPR1 = {c=4q+1+2*half}, M = lane%16.
// B (4x16 f32): same K striping, N = lane%16.
// C/D (16x16 f32): VGPR r holds (M = r + 8*half, N = lane%16).
// ---------------------------------------------------------------------------
__global__ __launch_bounds__(128)
void transform_wmma_full(const float* __restrict__ x,
                         const float* __restrict__ weight,
                         float* __restrict__ x_all) {
  const int lane = threadIdx.x;        // 0..31
  const int g    = threadIdx.y;        // 0..3 (one wave per group)
  const int half = lane >> 4;          // 0 | 1
  const int l    = lane & 15;
  const int n0   = blockIdx.x * 16;

  // A fragments: contiguous float2 loads, exactly the ISA lane layout.
  const float* ap = x + (size_t)(n0 + l) * IN_C + g * 16;
  v2f a0 = *(const v2f*)(ap + 0  + 2 * half);
  v2f a1 = *(const v2f*)(ap + 4  + 2 * half);
  v2f a2 = *(const v2f*)(ap + 8  + 2 * half);
  v2f a3 = *(const v2f*)(ap + 12 + 2 * half);

  // weight flat layout: idx = ((g*16 + c)*16 + d)*27 + k   (k innermost)
  int wb0, wb1, wb2, wb3;
  {
    int c0 = 0 + 2 * half, c1 = 4 + 2 * half, c2 = 8 + 2 * half, c3 = 12 + 2 * half;
    wb0 = ((g * 16 + c0) * 16 + l) * NK;
    wb1 = ((g * 16 + c1) * 16 + l) * NK;
    wb2 = ((g * 16 + c2) * 16 + l) * NK;
    wb3 = ((g * 16 + c3) * 16 + l) * NK;
  }

  // Per-lane output base: row m = n0 + r + 8*half, channel d = l.
  // Row step per r is NK*OUT_C floats (constant) -> immediate store offsets.
  float* ob = x_all + ((size_t)(n0 + 8 * half) * NK) * OUT_C + g * 16 + l;

  for (int k = 0; k < NK; ++k) {
    v2f b0, b1, b2, b3;
    b0.x = weight[wb0 + k]; b0.y = weight[wb0 + CSTRIDE + k];
    b1.x = weight[wb1 + k]; b1.y = weight[wb1 + CSTRIDE + k];
    b2.x = weight[wb2 + k]; b2.y = weight[wb2 + CSTRIDE + k];
    b3.x = weight[wb3 + k]; b3.y = weight[wb3 + CSTRIDE + k];

    v8f acc = {};
    // 8 args: (neg_a, A, neg_b, B, c_mod, C, reuse_a, reuse_b)
    acc = __builtin_amdgcn_wmma_f32_16x16x4_f32(false, a0, false, b0, (short)0, acc, false, false);
    acc = __builtin_amdgcn_wmma_f32_16x16x4_f32(false, a1, false, b1, (short)0, acc, false, false);
    acc = __builtin_amdgcn_wmma_f32_16x16x4_f32(false, a2, false, b2, (short)0, acc, false, false);
    acc = __builtin_amdgcn_wmma_f32_16x16x4_f32(false, a3, false, b3, (short)0, acc, false, false);

    // 8 coalesced b32 stores (lanes 0..15 / 16..31 each cover a contiguous
    // 64B line per r); compiler emits one s_clause of stores.
    float* op = ob + (size_t)k * OUT_C;
#pragma unroll
    for (int r = 0; r < 8; ++r)
      op[(size_t)r * (NK * OUT_C)] = acc[r];
  }
}

// ---------------------------------------------------------------------------
// Phase 1 tail: remaining (n % 16) nodes, plain scalar math (cold path).
// One thread per (node, k, channel) output element.
// ---------------------------------------------------------------------------
__global__ __launch_bounds__(256)
void transform_tail(const float* __restrict__ x,
                    const float* __restrict__ weight,
                    float* __restrict__ x_all,
                    int start, int n) {
  int idx   = blockIdx.x * blockDim.x + threadIdx.x;
  int total = (n - start) * NK * OUT_C;
  if (idx >= total) return;

  int node = start + idx / (NK * OUT_C);
  int rem  = idx % (NK * OUT_C);
  int k    = rem / OUT_C;
  int dch  = rem % OUT_C;
  int g    = dch >> 4;
  int d    = dch & 15;

  const float* xp = x + (size_t)node * IN_C + g * 16;
  float s = 0.f;
#pragma unroll
  for (int c = 0; c < 16; ++c)
    s += xp[c] * weight[((g * 16 + c) * 16 + d) * NK + k];

  x_all[(size_t)start * NK * OUT_C + idx] = s;
}

// ---------------------------------------------------------------------------
// out[n][64] = bias[d]
// ---------------------------------------------------------------------------
__global__ __launch_bounds__(256)
void bias_init(float* __restrict__ out, const float* __restrict__ bias, int total) {
  int idx = blockIdx.x * blockDim.x + threadIdx.x;
  if (idx < total) out[idx] = bias[idx & (OUT_C - 1)];
}

// ---------------------------------------------------------------------------
// Phase 2: one wave per edge. 256B coalesced gather of x_all[j][k][:] with a
// non-temporal hint (working set 691MB >> L2; keep L2 for the atomic target),
// then two f32 atomics per lane into out[i] (25.6MB, L2-resident).
// ---------------------------------------------------------------------------
__global__ __launch_bounds__(256)
void edge_scatter(const float* __restrict__ x_all,
                  const int* __restrict__ ei, const int* __restrict__ ej,
                  const int* __restrict__ ek,
                  float* __restrict__ out, int n_edges) {
  long gid = (long)blockIdx.x * blockDim.x + threadIdx.x;
  long e   = gid >> 5;
  int lane = (int)(gid & 31);
  if (e >= n_edges) return;

  int i = ei[e], j = ej[e], k = ek[e];
  const v2f* src = (const v2f*)(x_all + ((size_t)j * NK + k) * OUT_C);
  v2f v = __builtin_nontemporal_load(src + lane);

  float* dst = out + (size_t)i * OUT_C + lane * 2;
  atomicAdd(dst + 0, v.x);
  atomicAdd(dst + 1, v.y);
}

// ---------------------------------------------------------------------------
// Fallback (only if workspace < 691 MB): fused per-edge compute, no x_all.
// ---------------------------------------------------------------------------
__global__ __launch_bounds__(256)
void edge_fused(const float* __restrict__ x, const float* __restrict__ weight,
                const int* __restrict__ ei, const int* __restrict__ ej,
                const int* __restrict__ ek,
                float* __restrict__ out, int n_edges) {
  long gid = (long)blockIdx.x * blockDim.x + threadIdx.x;
  long e   = gid >> 5;
  int lane = (int)(gid & 31);
  if (e >= n_edges) return;

  int i = ei[e], j = ej[e], k = ek[e];
  int d0 = lane * 2;           // two output channels per lane, same group
  int g  = d0 >> 4;
  int dd = d0 & 15;

  const float* xj = x + (size_t)j * IN_C + g * 16;
  int base = (g * 16) * 16 * NK + k;   // + (c*16 + d)*NK
  float s0 = 0.f, s1 = 0.f;
#pragma unroll
  for (int c = 0; c < 16; ++c) {
    float xv = xj[c];
    s0 += xv * weight[base + (c * 16 + dd    ) * NK];
    s1 += xv * weight[base + (c * 16 + dd + 1) * NK];
  }
  float* dst = out + (size_t)i * OUT_C + d0;
  atomicAdd(dst + 0, s0);
  atomicAdd(dst + 1, s1);
}

// ---------------------------------------------------------------------------
extern "C" void kernel_launch(void* const* d_in, const int* in_sizes, int n_in,
                              void* d_out, int out_size, void* d_ws, size_t ws_size,
                              hipStream_t stream) {
  const float* x      = (const float*)d_in[0];
  const int*   ei     = (const int*)  d_in[1];
  const int*   ej     = (const int*)  d_in[2];
  const int*   ek     = (const int*)  d_in[3];
  // d_in[4] = n (scalar on device); derive n from sizes instead.
  const float* weight = (const float*)d_in[5];
  const float* bias   = (const float*)d_in[6];
  float* out = (float*)d_out;

  const int n       = in_sizes[0] / IN_C;
  const int n_edges = in_sizes[1];

  // out = bias (must precede atomics; stream ordering guarantees it)
  {
    int total = n * OUT_C;
    bias_init<<<(total + 255) / 256, 256, 0, stream>>>(out, bias, total);
  }

  const size_t need = (size_t)n * NK * OUT_C * sizeof(float);
  if (ws_size >= need) {
    float* x_all = (float*)d_ws;

    const int n_full = n / 16;          // full 16-node tiles (branch-free)
    if (n_full > 0) {
      dim3 blk(32, NG);                 // 4 waves/block, one per group
      transform_wmma_full<<<dim3(n_full), blk, 0, stream>>>(x, weight, x_all);
    }
    const int start = n_full * 16;      // remainder nodes (none for n=100000)
    if (start < n) {
      int total = (n - start) * NK * OUT_C;
      transform_tail<<<(total + 255) / 256, 256, 0, stream>>>(x, weight, x_all,
                                                              start, n);
    }

    long threads = (long)n_edges * 32;  // one wave per edge
    edge_scatter<<<(unsigned)((threads + 255) / 256), 256, 0, stream>>>(
        x_all, ei, ej, ek, out, n_edges);
  } else {
    long threads = (long)n_edges * 32;
    edge_fused<<<(unsigned)((threads + 255) / 256), 256, 0, stream>>>(
        x, weight, ei, ej, ek, out, n_edges);
  }
}